// BackboneTorsionWholePoseScoringModule_30313879176016
// MI455X (gfx1250) — compile-verified
//
#include <hip/hip_runtime.h>
#include <stdint.h>

#define N_POSES 256
#define N_BLOCKS 1024
#define N_ATS 15
#define NBIN 36

typedef __attribute__((ext_vector_type(2))) float v2f;
typedef __attribute__((ext_vector_type(8))) float v8f;

// ---- LDS layout (dword offsets; every sub-array is a multiple of 4 dwords, so 16B aligned) ----
#define OFF_TA   0                   // bt_backbone_torsion_atoms: 64*3*4*3 = 2304
#define OFF_DSC  (OFF_TA + 2304)     // bt_atom_downstream_of_conn: 64*2*4 = 512
#define OFF_RTAB (OFF_DSC + 512)     // bt_rama_table: 64*2 = 128
#define OFF_UPC  (OFF_RTAB + 128)    // bt_upper_conn_ind: 64
#define OFF_PRO  (OFF_UPC + 64)      // bt_is_pro: 64
#define OFF_RP   (OFF_PRO + 64)      // rama_table_params: 40*4 = 160
#define OFF_OP   (OFF_RP + 160)      // omega_table_params: 2*4 = 8
#define OFF_OM   (OFF_OP + 8)        // omega_tables: 2*36 = 72
#define OFF_CONN (OFF_OM + 72)       // conn for this pose: 1024*2*2 = 4096
#define OFF_OFFS (OFF_CONN + 4096)   // coord offsets: 1024
#define OFF_BT   (OFF_OFFS + 1024)   // block types: 1024
#define LDS_DWORDS (OFF_BT + 1024)   // 9456 dwords = 37824 B

__device__ __forceinline__ int imin(int a, int b) { return a < b ? a : b; }
__device__ __forceinline__ int imax(int a, int b) { return a > b ? a : b; }
__device__ __forceinline__ int mod36(int v) { int m = v % 36; return m < 0 ? m + 36 : m; }

// ---- CDNA5 async global->LDS copy (GV mode: per-lane 64-bit global addr + 32-bit LDS addr) ----
__device__ __forceinline__ void async_b128(uint32_t lds_byte_addr, const void* gaddr) {
    asm volatile("global_load_async_to_lds_b128 %0, %1, off"
                 :: "v"(lds_byte_addr), "v"(gaddr)
                 : "memory");
}

__device__ __forceinline__ void stage(int* smem, const int* gsrc, int base_dw, int ndw, int tid) {
    uint32_t lds_base = (uint32_t)(uintptr_t)(smem + base_dw);
    for (int i = tid * 4; i < ndw; i += 1024 * 4) {
        async_b128(lds_base + (uint32_t)i * 4u, gsrc + i);
    }
}

struct F3 { float x, y, z; };
__device__ __forceinline__ F3 f3sub(F3 a, F3 b) { return {a.x - b.x, a.y - b.y, a.z - b.z}; }
__device__ __forceinline__ F3 f3cross(F3 a, F3 b) {
    return {a.y * b.z - a.z * b.y, a.z * b.x - a.x * b.z, a.x * b.y - a.y * b.x};
}
__device__ __forceinline__ float f3dot(F3 a, F3 b) { return a.x * b.x + a.y * b.y + a.z * b.z; }

__device__ __forceinline__ void crw(float t, float w[4]) {
    float t2 = t * t, t3 = t2 * t;
    w[0] = -0.5f * t3 + t2 - 0.5f * t;
    w[1] =  1.5f * t3 - 2.5f * t2 + 1.0f;
    w[2] = -1.5f * t3 + 2.0f * t2 + 0.5f * t;
    w[3] =  0.5f * t3 - 0.5f * t2;
}

// Exact 32-lane wave sum via V_WMMA_F32_16X16X4_F32.
// A (16x4 f32, documented layout): lanes 0-15 v0->K0,v1->K1 of row M=lane;
// lanes 16-31 v0->K2,v1->K3 of row M=lane-16.  Put x in both -> A[m][0..1]=x[m], A[m][2..3]=x[m+16].
// B = all ones (layout independent)  =>  D[m][n] = 2*(x[m] + x[m+16]).
// D layout: vgpr j = row M=j (lanes 0-15) / M=j+8 (lanes 16-31).  Sum 8 vgprs, fold halves, halve.
__device__ __forceinline__ float wave_sum_wmma(float x) {
    v2f a; a.x = x; a.y = x;
    v2f ones; ones.x = 1.0f; ones.y = 1.0f;
    v8f c = {};
    v8f d = __builtin_amdgcn_wmma_f32_16x16x4_f32(false, a, false, ones, (short)0, c, false, false);
    float s = d[0] + d[1] + d[2] + d[3] + d[4] + d[5] + d[6] + d[7];
    s += __shfl_xor(s, 16, 32);
    return s * 0.5f;
}

__global__ __launch_bounds__(1024, 1)
void backbone_torsion_kernel(const float* __restrict__ coords,
                             const int*   __restrict__ g_off,
                             const int*   __restrict__ g_bt,
                             const int*   __restrict__ g_conn,
                             const int*   __restrict__ g_dsc,
                             const int*   __restrict__ g_rtab,
                             const int*   __restrict__ g_upc,
                             const int*   __restrict__ g_pro,
                             const int*   __restrict__ g_ta,
                             const float* __restrict__ g_rama,
                             const float* __restrict__ g_om,
                             const float* __restrict__ g_rp,
                             const float* __restrict__ g_op,
                             float*       __restrict__ out)
{
    __shared__ int   smem[LDS_DWORDS];
    __shared__ float s_red[64];

    const int tid = threadIdx.x;
    const int p   = blockIdx.x;

    // ---- async-stage all metadata into LDS (CDNA5 async path, ASYNCcnt tracked) ----
    stage(smem, g_ta,                 OFF_TA,   2304, tid);
    stage(smem, g_dsc,                OFF_DSC,   512, tid);
    stage(smem, g_rtab,               OFF_RTAB,  128, tid);
    stage(smem, g_upc,                OFF_UPC,    64, tid);
    stage(smem, g_pro,                OFF_PRO,    64, tid);
    stage(smem, (const int*)g_rp,     OFF_RP,    160, tid);
    stage(smem, (const int*)g_op,     OFF_OP,      8, tid);
    stage(smem, (const int*)g_om,     OFF_OM,     72, tid);
    stage(smem, g_conn + p * 4096,    OFF_CONN, 4096, tid);
    stage(smem, g_off  + p * 1024,    OFF_OFFS, 1024, tid);
    stage(smem, g_bt   + p * 1024,    OFF_BT,   1024, tid);
    asm volatile("s_wait_asynccnt 0x0" ::: "memory");
    __syncthreads();

    const int*   s_ta   = smem + OFF_TA;
    const int*   s_dsc  = smem + OFF_DSC;
    const int*   s_rtab = smem + OFF_RTAB;
    const int*   s_upc  = smem + OFF_UPC;
    const int*   s_pro  = smem + OFF_PRO;
    const float* s_rp   = (const float*)(smem + OFF_RP);
    const float* s_op   = (const float*)(smem + OFF_OP);
    const float* s_om   = (const float*)(smem + OFF_OM);
    const int*   s_conn = smem + OFF_CONN;
    const int*   s_off  = smem + OFF_OFFS;
    const int*   s_bt   = smem + OFF_BT;

    const int b      = tid;
    const int btraw  = s_bt[b];
    const bool real  = btraw >= 0;
    const int bt     = imax(btraw, 0);
    const int off_b  = s_off[b];
    const float* pose_coords = coords + (size_t)p * (N_BLOCKS * N_ATS) * 3;

    float dih[3];
    bool  tv[3];
    for (int t = 0; t < 3; ++t) {
        bool allv = real;
        F3 pt[4];
        for (int k = 0; k < 4; ++k) {
            const int base = bt * 36 + t * 12 + k * 3;
            const int a  = s_ta[base + 0];
            const int ci = s_ta[base + 1];
            const int nb = s_ta[base + 2];
            int g;
            if (a >= 0) {
                g = off_b + a;                     // intra-residue atom
            } else {
                const int cc        = imin(imax(ci, 0), 1);
                const int nbr_block = s_conn[(b * 2 + cc) * 2 + 0];
                const int nbr_conn  = s_conn[(b * 2 + cc) * 2 + 1];
                const int nbc       = imax(nbr_block, 0);
                const int nbt       = imax(s_bt[nbc], 0);
                const int ds        = s_dsc[nbt * 8 + imin(imax(nbr_conn, 0), 1) * 4 + imin(imax(nb, 0), 3)];
                g = s_off[nbc] + ds;
                allv = allv && (ci >= 0) && (nbr_block >= 0) && (ds >= 0);
            }
            const float* cp = pose_coords + (size_t)imax(g, 0) * 3;
            pt[k].x = __ldg(cp + 0);
            pt[k].y = __ldg(cp + 1);
            pt[k].z = __ldg(cp + 2);
        }
        tv[t] = allv;

        const F3 b1 = f3sub(pt[1], pt[0]);
        const F3 b2 = f3sub(pt[2], pt[1]);
        const F3 b3 = f3sub(pt[3], pt[2]);
        const F3 n1 = f3cross(b1, b2);
        const F3 n2 = f3cross(b2, b3);
        const float nrm = sqrtf(f3dot(b2, b2));
        const F3 b2n = {b2.x / nrm, b2.y / nrm, b2.z / nrm};
        const F3 m1 = f3cross(n1, b2n);
        dih[t] = atan2f(f3dot(m1, n2), f3dot(n1, n2));
    }

    // ---- rama (bicubic Catmull-Rom on phi/psi) ----
    const int uc         = imin(imax(s_upc[bt], 0), 1);
    const int next_block = s_conn[(b * 2 + uc) * 2 + 0];
    const int next_bt    = imax(s_bt[imax(next_block, 0)], 0);
    const int ipn        = (next_block >= 0) ? imin(imax(s_pro[next_bt], 0), 1) : 0;
    const int rind       = imin(imax(s_rtab[bt * 2 + ipn], 0), 39);
    const float* rp = s_rp + rind * 4;
    const float x = (dih[0] - rp[0]) / rp[2];
    const float y = (dih[1] - rp[1]) / rp[3];
    const float xf = floorf(x), yf = floorf(y);
    const float fx = x - xf,   fy = y - yf;
    const int ix0 = (int)xf, iy0 = (int)yf;
    float wx[4], wy[4];
    crw(fx, wx); crw(fy, wy);
    const float* tab = g_rama + rind * (NBIN * NBIN);
    int iys[4];
    for (int j = 0; j < 4; ++j) iys[j] = mod36(iy0 - 1 + j);
    float e_r = 0.0f;
    for (int i = 0; i < 4; ++i) {
        const float* row = tab + mod36(ix0 - 1 + i) * NBIN;
        float acc = wy[0] * __ldg(row + iys[0]) + wy[1] * __ldg(row + iys[1])
                  + wy[2] * __ldg(row + iys[2]) + wy[3] * __ldg(row + iys[3]);
        e_r += wx[i] * acc;
    }
    const float rama_e = (tv[0] && tv[1]) ? e_r : 0.0f;

    // ---- omega (1-D Catmull-Rom) ----
    const int om_ind = imin(imax(s_pro[bt], 0), 1);
    const float* op = s_op + om_ind * 4;
    const float z  = (dih[2] - op[0]) / op[1];
    const float zf = floorf(z);
    const float fz = z - zf;
    const int iz0 = (int)zf;
    float wz[4];
    crw(fz, wz);
    const float* otab = s_om + om_ind * NBIN;
    float e_o = 0.0f;
    for (int i = 0; i < 4; ++i) e_o += wz[i] * otab[mod36(iz0 - 1 + i)];
    const float om_e = tv[2] ? e_o : 0.0f;

    // ---- reduction: WMMA wave sum, then LDS across 32 waves ----
    const float rsum = wave_sum_wmma(rama_e);
    const float osum = wave_sum_wmma(om_e);
    const int wid = tid >> 5;
    if ((tid & 31) == 0) { s_red[wid] = rsum; s_red[32 + wid] = osum; }
    __syncthreads();
    if (tid == 0) {
        float r = 0.0f, o = 0.0f;
        for (int w = 0; w < 32; ++w) { r += s_red[w]; o += s_red[32 + w]; }
        out[p] = r;
        out[N_POSES + p] = o;
    }
}

extern "C" void kernel_launch(void* const* d_in, const int* in_sizes, int n_in,
                              void* d_out, int out_size, void* d_ws, size_t ws_size,
                              hipStream_t stream) {
    (void)in_sizes; (void)n_in; (void)out_size; (void)d_ws; (void)ws_size;
    const float* coords = (const float*)d_in[0];
    const int*   g_off  = (const int*)d_in[1];
    const int*   g_bt   = (const int*)d_in[2];
    const int*   g_conn = (const int*)d_in[3];
    const int*   g_dsc  = (const int*)d_in[4];
    const int*   g_rtab = (const int*)d_in[5];
    const int*   g_upc  = (const int*)d_in[6];
    const int*   g_pro  = (const int*)d_in[7];
    const int*   g_ta   = (const int*)d_in[8];
    const float* g_rama = (const float*)d_in[9];
    const float* g_om   = (const float*)d_in[10];
    const float* g_rp   = (const float*)d_in[11];
    const float* g_op   = (const float*)d_in[12];
    float* out = (float*)d_out;

    hipLaunchKernelGGL(backbone_torsion_kernel, dim3(N_POSES), dim3(N_BLOCKS), 0, stream,
                       coords, g_off, g_bt, g_conn, g_dsc, g_rtab, g_upc, g_pro, g_ta,
                       g_rama, g_om, g_rp, g_op, out);
}